// SAGELayer_28647431864953
// MI455X (gfx1250) — compile-verified
//
#include <hip/hip_runtime.h>

// SAGEConv (mean aggregation) for MI455X / gfx1250.
//   out = (segment_mean(x[src], dst)) @ W_l + x @ W_r + b
// N = 50000 nodes, F = 128 features, E = 800000 edges.

#define F 128
#define LDA 132   // LDS row pitch in floats: 132 % 64 == 4 -> conflict-free b64 reads

typedef float v2f __attribute__((ext_vector_type(2)));
typedef float v8f __attribute__((ext_vector_type(8)));

// ---------------------------------------------------------------------------
// Kernel 1: zero the workspace (summed[N*F] followed by deg[N]).
// ---------------------------------------------------------------------------
__global__ void sage_zero_kernel(float* __restrict__ p, int n) {
    int i = blockIdx.x * blockDim.x + threadIdx.x;
    if (i < n) p[i] = 0.0f;
}

// ---------------------------------------------------------------------------
// Kernel 2: edge scatter. One wave (32 lanes) per edge.
// Each lane loads float4 of x[src] (global_load_b128, 512B contiguous row)
// and does 4 global f32 atomic adds into summed[dst]. Lane 0 counts degree.
// summed (25.6 MB) is L2-resident, so atomics resolve in L2.
// ---------------------------------------------------------------------------
__global__ void sage_scatter_kernel(const float* __restrict__ x,
                                    const long long* __restrict__ src,
                                    const long long* __restrict__ dst,
                                    float* __restrict__ summed,
                                    float* __restrict__ deg,
                                    int E) {
    int gid  = blockIdx.x * blockDim.x + threadIdx.x;
    int e    = gid >> 5;
    int lane = threadIdx.x & 31;
    if (e >= E) return;

    int s = (int)src[e];
    int d = (int)dst[e];

    float4 v = ((const float4*)(x + (size_t)s * F))[lane];
    float* o = summed + (size_t)d * F + lane * 4;
    atomicAdd(o + 0, v.x);
    atomicAdd(o + 1, v.y);
    atomicAdd(o + 2, v.z);
    atomicAdd(o + 3, v.w);
    if (lane == 0) atomicAdd(deg + d, 1.0f);
}

// ---------------------------------------------------------------------------
// Kernel 3: fused dual GEMM + bias using V_WMMA_F32_16X16X4_F32.
//   out[16 rows x 128 cols per block] = mean @ W_l + x @ W_r + b
// Block = 256 threads = 8 waves; wave w computes the 16x16 tile at cols
// [16w, 16w+16). A panels (mean, x) are staged in LDS; W_l/W_r stream from
// L2 (64 KB each, fully resident).
//
// Fragment layouts (ISA 7.12.2, 32-bit, wave32):
//   A 16x4 : lane l -> M = l&15, VGPR0/1 = K = (l>>4)*2 + {0,1}
//   B 4x16 : lane l -> N = l&15, VGPR0/1 = K = (l>>4)*2 + {0,1}
//   C/D    : lane l -> N = l&15, VGPR v -> M = v + (l>=16 ? 8 : 0)
// ---------------------------------------------------------------------------
__global__ void __launch_bounds__(256)
sage_gemm_kernel(const float* __restrict__ x,
                 const float* __restrict__ summed,
                 const float* __restrict__ deg,
                 const float* __restrict__ Wl,
                 const float* __restrict__ Wr,
                 const float* __restrict__ bias,
                 float* __restrict__ out,
                 int N) {
    __shared__ float sMean[16 * LDA];
    __shared__ float sX[16 * LDA];
    __shared__ float sInv[16];

    const int tid  = threadIdx.x;
    const int row0 = blockIdx.x * 16;

    // Per-row 1/max(deg,1)
    if (tid < 16) {
        int node = row0 + tid;
        float d  = (node < N) ? deg[node] : 1.0f;
        sInv[tid] = 1.0f / fmaxf(d, 1.0f);
    }
    __syncthreads();

    // Stage 16x128 panels of mean and x into LDS (float4 global loads).
    // 512 float4 elements per panel, 256 threads -> 2 each.
    for (int i = tid; i < 16 * (F / 4); i += 256) {
        int r    = i >> 5;        // row 0..15
        int c4   = i & 31;        // float4 column 0..31
        int node = row0 + r;
        int nclamp = node < N ? node : (N - 1);
        float4 sv = ((const float4*)(summed + (size_t)nclamp * F))[c4];
        float4 xv = ((const float4*)(x      + (size_t)nclamp * F))[c4];
        float inv = sInv[r];
        float* pm = &sMean[r * LDA + c4 * 4];
        float* px = &sX[r * LDA + c4 * 4];
        pm[0] = sv.x * inv; pm[1] = sv.y * inv; pm[2] = sv.z * inv; pm[3] = sv.w * inv;
        px[0] = xv.x;       px[1] = xv.y;       px[2] = xv.z;       px[3] = xv.w;
    }
    __syncthreads();

    const int wave  = tid >> 5;            // 0..7 -> column tile
    const int lane  = tid & 31;
    const int m     = lane & 15;           // A row / C-D column index base
    const int khalf = (lane >> 4) * 2;     // K offset: lanes 16..31 take K+2
    const int n     = (wave << 4) + (lane & 15);   // output column 0..127

    v8f acc = {};

    // mean @ W_l
    #pragma unroll 8
    for (int k0 = 0; k0 < F; k0 += 4) {
        v2f a = *(const v2f*)&sMean[m * LDA + k0 + khalf];
        v2f bfrag;
        bfrag.x = Wl[(size_t)(k0 + khalf)     * F + n];
        bfrag.y = Wl[(size_t)(k0 + khalf + 1) * F + n];
        acc = __builtin_amdgcn_wmma_f32_16x16x4_f32(
            false, a, false, bfrag, (short)0, acc, false, false);
    }
    // + x @ W_r
    #pragma unroll 8
    for (int k0 = 0; k0 < F; k0 += 4) {
        v2f a = *(const v2f*)&sX[m * LDA + k0 + khalf];
        v2f bfrag;
        bfrag.x = Wr[(size_t)(k0 + khalf)     * F + n];
        bfrag.y = Wr[(size_t)(k0 + khalf + 1) * F + n];
        acc = __builtin_amdgcn_wmma_f32_16x16x4_f32(
            false, a, false, bfrag, (short)0, acc, false, false);
    }

    // Bias + store per C/D layout.
    float bv = bias[n];
    int mbase = (lane >> 4) * 8;
    #pragma unroll
    for (int v = 0; v < 8; ++v) {
        int row = row0 + mbase + v;
        if (row < N) out[(size_t)row * F + n] = acc[v] + bv;
    }
}

// ---------------------------------------------------------------------------
// Launcher
// ---------------------------------------------------------------------------
extern "C" void kernel_launch(void* const* d_in, const int* in_sizes, int n_in,
                              void* d_out, int out_size, void* d_ws, size_t ws_size,
                              hipStream_t stream) {
    const float*     x   = (const float*)d_in[0];
    const long long* ei  = (const long long*)d_in[1];   // [2, E] int64
    const float*     Wl  = (const float*)d_in[2];
    const float*     Wr  = (const float*)d_in[3];
    const float*     b   = (const float*)d_in[4];
    float*           out = (float*)d_out;

    const int N = in_sizes[0] / F;       // 50000
    const int E = in_sizes[1] / 2;       // 800000

    const long long* src = ei;           // row 0
    const long long* dst = ei + E;       // row 1

    float* summed = (float*)d_ws;        // N*F floats
    float* deg    = summed + (size_t)N * F;  // N floats

    // 1) zero summed + deg
    {
        int n = N * F + N;
        int blocks = (n + 255) / 256;
        sage_zero_kernel<<<blocks, 256, 0, stream>>>(summed, n);
    }
    // 2) edge scatter: one wave per edge -> E*32 threads
    {
        long long threads = (long long)E * 32;
        int blocks = (int)((threads + 255) / 256);
        sage_scatter_kernel<<<blocks, 256, 0, stream>>>(x, src, dst, summed, deg, E);
    }
    // 3) fused dual WMMA GEMM + bias: one block per 16 rows
    {
        int blocks = (N + 15) / 16;
        sage_gemm_kernel<<<blocks, 256, 0, stream>>>(x, summed, deg, Wl, Wr, b, out, N);
    }
}